// Encoder_63239098466535
// MI455X (gfx1250) — compile-verified
//
#include <hip/hip_runtime.h>
#include <math.h>

typedef __attribute__((ext_vector_type(2))) float v2f;
typedef __attribute__((ext_vector_type(8))) float v8f;

#define BN_EPS 1e-5f

// ---- CDNA5 fp32 WMMA: D(16x16) = A(16x4) * B(4x16) + C, full fp32 precision ----
__device__ __forceinline__ v8f wmma4(v2f a, v2f b, v8f c) {
  // 8 args: (neg_a, A, neg_b, B, c_mod, C, reuse_a, reuse_b)
  return __builtin_amdgcn_wmma_f32_16x16x4_f32(false, a, false, b, (short)0, c,
                                               false, false);
}

__device__ __forceinline__ bool any32(bool p) {
#if __has_builtin(__builtin_amdgcn_ballot_w32)
  return __builtin_amdgcn_ballot_w32(p) != 0u;
#else
  return true;
#endif
}

__device__ __forceinline__ void atomAddAgent(float* p, float v) {
  __hip_atomic_fetch_add(p, v, __ATOMIC_RELAXED, __HIP_MEMORY_SCOPE_AGENT);
}
__device__ __forceinline__ void atomAddWG(float* p, float v) {
  __hip_atomic_fetch_add(p, v, __ATOMIC_RELAXED, __HIP_MEMORY_SCOPE_WORKGROUP);
}

// ---------------- init: stats floats + partition counters ----------------
__global__ void k_zero_ws(float* stats, int* cnt) {
  int t = threadIdx.x;
  if (t < 256) stats[t] = 0.0f;
  if (t < 32) cnt[t] = 0;
}

__global__ void k_init_bias(float* __restrict__ x, long total, int cmask,
                            const float* __restrict__ b) {
  long i = (long)blockIdx.x * blockDim.x + threadIdx.x;
  long stride = (long)gridDim.x * blockDim.x;
  for (; i < total; i += stride) x[i] = b[i & cmask];
}

// ---------------- k-partition: histogram / prefix / rank-scatter ----------------
__global__ __launch_bounds__(256) void k_hist(const int* __restrict__ kidx,
                                              int Np, int* __restrict__ tot) {
  __shared__ int h[8];
  const int t = threadIdx.x;
  if (t < 8) h[t] = 0;
  __syncthreads();
  int i = blockIdx.x * 256 + t;
  if (i < Np) atomicAdd(&h[kidx[i]], 1);
  __syncthreads();
  if (t < 8 && h[t] > 0) atomicAdd(&tot[t], h[t]);
}

__global__ void k_prefix(int* __restrict__ cnt) {
  // cnt layout: tot0[0..8) cur0[8..16) tot1[16..24) cur1[24..32)
  if (threadIdx.x == 0 && blockIdx.x == 0) {
    int run = 0;
    for (int k = 0; k < 8; ++k) { cnt[8 + k] = run; run += cnt[k]; }
    run = 0;
    for (int k = 0; k < 8; ++k) { cnt[24 + k] = run; run += cnt[16 + k]; }
  }
}

__global__ __launch_bounds__(256) void k_scatter_perm(
    const int* __restrict__ kidx, int Np, int* __restrict__ cur,
    int* __restrict__ perm) {
  __shared__ int h[8];
  __shared__ int base[8];
  const int t = threadIdx.x;
  if (t < 8) h[t] = 0;
  __syncthreads();
  int i = blockIdx.x * 256 + t;
  int kid = -1;
  if (i < Np) {
    kid = kidx[i];
    atomicAdd(&h[kid], 1);
  }
  __syncthreads();
  if (t < 8) {
    base[t] = h[t] ? atomicAdd(&cur[t], h[t]) : 0;
    h[t] = 0;
  }
  __syncthreads();
  if (i < Np) {
    int r = atomicAdd(&h[kid], 1);
    perm[base[kid] + r] = i;
  }
}

// ---------------- conv layer 0: Cin=4, Cout=32 ----------------
// One wave = 16 points (k-sorted via perm). A-frag (16x4 f32): lane<16 ->
// M=lane, K={0,1}; lane>=16 -> M=lane-16, K={2,3}. Tap mask zeroes A rows
// with kidx != k; sorted tiles make the ballot skip ~7 of 8 taps.
__global__ __launch_bounds__(256) void k_conv0(
    const float* __restrict__ feats, const float* __restrict__ W0,
    const int* __restrict__ kidx, const int* __restrict__ parent,
    const int* __restrict__ perm, float* __restrict__ out, int N, int ntiles) {
  const int lane = threadIdx.x & 31;
  const int wave = threadIdx.x >> 5;
  const int tile = blockIdx.x * (blockDim.x >> 5) + wave;
  if (tile >= ntiles) return;  // wave-uniform

  const int tbase = tile * 16;
  const int half = lane >> 4;
  const int col = lane & 15;
  const int K0 = half * 2;

  int idxA = tbase + col;
  int idxAc = idxA < N ? idxA : N - 1;
  int rowA = perm ? perm[idxAc] : idxAc;
  int kid = kidx[rowA];
  if (idxA >= N) kid = -1;
  v2f a = *(const v2f*)(feats + (size_t)rowA * 4 + K0);

  v8f c0 = {0.f, 0.f, 0.f, 0.f, 0.f, 0.f, 0.f, 0.f};
  v8f c1 = c0;
  const v2f zz = {0.f, 0.f};

#pragma unroll
  for (int k = 0; k < 8; ++k) {
    bool m = (kid == k);
    if (any32(m)) {  // uniform branch: EXEC stays all-ones for WMMA
      v2f ak = m ? a : zz;
      const float* wb = W0 + k * 128 + K0 * 32 + col;  // W0[k][cin][cout]
      v2f b;
      b.x = wb[0];  b.y = wb[32];   // cols 0..15
      c0 = wmma4(ak, b, c0);
      b.x = wb[16]; b.y = wb[48];   // cols 16..31
      c1 = wmma4(ak, b, c1);
    }
  }

  // scatter: C vgpr r holds row M=half*8+r, column N=col
#pragma unroll
  for (int r = 0; r < 8; ++r) {
    int idxS = tbase + half * 8 + r;
    if (idxS < N) {
      int rowS = perm ? perm[idxS] : idxS;
      int p = parent[rowS];
      float* dst = out + (size_t)p * 32 + col;
      atomAddAgent(dst + 0, c0[r]);
      atomAddAgent(dst + 16, c1[r]);
    }
  }
}

// ---------------- conv layer 1: Cin=32, Cout=64 ----------------
__global__ __launch_bounds__(256) void k_conv1(
    const float* __restrict__ feats, const float* __restrict__ W1,
    const int* __restrict__ kidx, const int* __restrict__ parent,
    const int* __restrict__ perm, float* __restrict__ out, int N, int ntiles) {
  const int lane = threadIdx.x & 31;
  const int wave = threadIdx.x >> 5;
  const int tile = blockIdx.x * (blockDim.x >> 5) + wave;
  if (tile >= ntiles) return;

  const int tbase = tile * 16;
  const int half = lane >> 4;
  const int col = lane & 15;
  const int K0 = half * 2;

  int idxA = tbase + col;
  int idxAc = idxA < N ? idxA : N - 1;
  int rowA = perm ? perm[idxAc] : idxAc;
  int kid = kidx[rowA];
  if (idxA >= N) kid = -1;

  v2f a[8];
#pragma unroll
  for (int kc = 0; kc < 8; ++kc)
    a[kc] = *(const v2f*)(feats + (size_t)rowA * 32 + kc * 4 + K0);

  const v8f z8 = {0.f, 0.f, 0.f, 0.f, 0.f, 0.f, 0.f, 0.f};
  v8f c[4] = {z8, z8, z8, z8};
  const v2f zz = {0.f, 0.f};

#pragma unroll
  for (int k = 0; k < 8; ++k) {
    bool m = (kid == k);
    if (any32(m)) {
#pragma unroll
      for (int kc = 0; kc < 8; ++kc) {
        v2f ak = m ? a[kc] : zz;
        // W1[k][cin][cout], cin = kc*4 + K0 (+1 for second K row)
        const float* wb = W1 + k * 2048 + (kc * 4 + K0) * 64 + col;
#pragma unroll
        for (int n = 0; n < 4; ++n) {
          v2f b;
          b.x = wb[n * 16];
          b.y = wb[n * 16 + 64];
          c[n] = wmma4(ak, b, c[n]);
        }
      }
    }
  }

#pragma unroll
  for (int r = 0; r < 8; ++r) {
    int idxS = tbase + half * 8 + r;
    if (idxS < N) {
      int rowS = perm ? perm[idxS] : idxS;
      int p = parent[rowS];
      float* dst = out + (size_t)p * 64 + col;
#pragma unroll
      for (int n = 0; n < 4; ++n) atomAddAgent(dst + n * 16, c[n][r]);
    }
  }
}

// ---------------- per-channel sum / sumsq reduction ----------------
// stride is a multiple of C, so each thread's channel is fixed.
__global__ __launch_bounds__(256) void k_reduce(const float* __restrict__ x,
                                                long total, int cmask,
                                                float* __restrict__ sum,
                                                float* __restrict__ sq) {
  __shared__ float ls[64];
  __shared__ float lq[64];
  const int t = threadIdx.x;
  const int C = cmask + 1;
  if (t < C) { ls[t] = 0.f; lq[t] = 0.f; }
  __syncthreads();

  long i0 = (long)blockIdx.x * blockDim.x + t;
  long stride = (long)gridDim.x * blockDim.x;
  const int c = (int)(i0 & cmask);
  float s = 0.f, q = 0.f;
  for (long i = i0; i < total; i += stride) {
    float v = x[i];
    s += v;
    q += v * v;
  }
  atomAddWG(&ls[c], s);
  atomAddWG(&lq[c], q);
  __syncthreads();
  if (t < C) {
    atomAddAgent(&sum[t], ls[t]);
    atomAddAgent(&sq[t], lq[t]);
  }
}

// ---------------- BatchNorm (batch stats) + ELU, in place ----------------
__global__ __launch_bounds__(256) void k_bn_elu(
    float* __restrict__ x, long total, int cmask, float inv_n,
    const float* __restrict__ sum, const float* __restrict__ sq,
    const float* __restrict__ gamma, const float* __restrict__ beta) {
  long i = (long)blockIdx.x * blockDim.x + threadIdx.x;
  long stride = (long)gridDim.x * blockDim.x;
  for (; i < total; i += stride) {
    int c = (int)(i & cmask);
    float mean = sum[c] * inv_n;
    float var = sq[c] * inv_n - mean * mean;
    float scale = gamma[c] * rsqrtf(var + BN_EPS);
    float v = (x[i] - mean) * scale + beta[c];
    x[i] = v > 0.f ? v : expm1f(v);
  }
}

extern "C" void kernel_launch(void* const* d_in, const int* in_sizes, int n_in,
                              void* d_out, int out_size, void* d_ws,
                              size_t ws_size, hipStream_t stream) {
  const float* feats = (const float*)d_in[0];
  const float* W0 = (const float*)d_in[1];
  const float* b0 = (const float*)d_in[2];
  const float* g0 = (const float*)d_in[3];
  const float* be0 = (const float*)d_in[4];
  const float* W1 = (const float*)d_in[5];
  const float* b1 = (const float*)d_in[6];
  const float* g1 = (const float*)d_in[7];
  const float* be1 = (const float*)d_in[8];
  const int* kidx1 = (const int*)d_in[9];
  const int* parent1 = (const int*)d_in[10];
  const int* kidx2 = (const int*)d_in[12];
  const int* parent2 = (const int*)d_in[13];

  const int N0 = in_sizes[0] / 4;   // input points
  const int N1 = in_sizes[11] / 3;  // coords1 rows
  const int N2 = in_sizes[14] / 3;  // coords2 rows

  float* out1 = (float*)d_out;            // [N2, 64]  (out_en1)
  float* out0 = out1 + (size_t)N2 * 64;   // [N1, 32]  (out_en0)

  // ws layout: [0,1024) stats floats; [1024,1152) partition counters;
  //            [2048, ...) perm0[N0], perm1[N1]
  char* wsb = (char*)d_ws;
  float* stats = (float*)wsb;
  int* cnt = (int*)(wsb + 1024);
  float* sum0 = stats;
  float* sq0 = stats + 32;
  float* sum1 = stats + 64;
  float* sq1 = stats + 128;
  int* tot0 = cnt;        // [0..8)
  int* cur0 = cnt + 8;    // [8..16)
  int* tot1 = cnt + 16;   // [16..24)
  int* cur1 = cnt + 24;   // [24..32)

  const bool use_perm =
      ws_size >= (size_t)2048 + (size_t)(N0 + (long)N1) * sizeof(int);
  int* perm0 = use_perm ? (int*)(wsb + 2048) : nullptr;
  int* perm1 = use_perm ? perm0 + N0 : nullptr;

  const long tot_e0 = (long)N1 * 32;
  const long tot_e1 = (long)N2 * 64;

  hipLaunchKernelGGL(k_zero_ws, dim3(1), dim3(256), 0, stream, stats, cnt);

  // ---- k-partition (independent of feature data; do it up front) ----
  if (use_perm) {
    const int nb0 = (N0 + 255) / 256;
    const int nb1 = (N1 + 255) / 256;
    hipLaunchKernelGGL(k_hist, dim3(nb0), dim3(256), 0, stream, kidx1, N0, tot0);
    hipLaunchKernelGGL(k_hist, dim3(nb1), dim3(256), 0, stream, kidx2, N1, tot1);
    hipLaunchKernelGGL(k_prefix, dim3(1), dim3(1), 0, stream, cnt);
    hipLaunchKernelGGL(k_scatter_perm, dim3(nb0), dim3(256), 0, stream, kidx1,
                       N0, cur0, perm0);
    hipLaunchKernelGGL(k_scatter_perm, dim3(nb1), dim3(256), 0, stream, kidx2,
                       N1, cur1, perm1);
  }

  int gb0 = (int)((tot_e0 + 255) / 256); if (gb0 > 4096) gb0 = 4096; if (gb0 < 1) gb0 = 1;
  int gb1 = (int)((tot_e1 + 255) / 256); if (gb1 > 4096) gb1 = 4096; if (gb1 < 1) gb1 = 1;
  hipLaunchKernelGGL(k_init_bias, dim3(gb0), dim3(256), 0, stream, out0, tot_e0, 31, b0);
  hipLaunchKernelGGL(k_init_bias, dim3(gb1), dim3(256), 0, stream, out1, tot_e1, 63, b1);

  // ---- layer 0 ----
  const int tiles0 = (N0 + 15) / 16;
  hipLaunchKernelGGL(k_conv0, dim3((tiles0 + 7) / 8), dim3(256), 0, stream,
                     feats, W0, kidx1, parent1, perm0, out0, N0, tiles0);
  hipLaunchKernelGGL(k_reduce, dim3(512), dim3(256), 0, stream, out0, tot_e0,
                     31, sum0, sq0);
  hipLaunchKernelGGL(k_bn_elu, dim3(gb0), dim3(256), 0, stream, out0, tot_e0,
                     31, 1.0f / (float)N1, sum0, sq0, g0, be0);

  // ---- layer 1 (input = post-BN/ELU out0) ----
  const int tiles1 = (N1 + 15) / 16;
  hipLaunchKernelGGL(k_conv1, dim3((tiles1 + 7) / 8), dim3(256), 0, stream,
                     out0, W1, kidx2, parent2, perm1, out1, N1, tiles1);
  hipLaunchKernelGGL(k_reduce, dim3(512), dim3(256), 0, stream, out1, tot_e1,
                     63, sum1, sq1);
  hipLaunchKernelGGL(k_bn_elu, dim3(gb1), dim3(256), 0, stream, out1, tot_e1,
                     63, 1.0f / (float)N2, sum1, sq1, g1, be1);
}